// NFFT_790273983130
// MI455X (gfx1250) — compile-verified
//
#include <hip/hip_runtime.h>
#include <math.h>

// NFFT(N=256x256, n=512x512, m=4), B=2, J=262144
#define NG   256
#define NOS  512
#define BATCH 2
#define JPTS 262144

typedef __attribute__((ext_vector_type(2))) float v2f;
typedef __attribute__((ext_vector_type(8))) float v8f;

__device__ __forceinline__ v8f wmma4(v2f a, v2f b, v8f c) {
  // D = A(16x4) * B(4x16) + C, full f32
  return __builtin_amdgcn_wmma_f32_16x16x4_f32(false, a, false, b, (short)0, c,
                                               false, false);
}

// I0(x) for x >= 3.75 (Abramowitz-Stegun 9.8.2), rel err ~1e-7 at x~18
__device__ __forceinline__ float bessel_i0_large(float x) {
  float t = 3.75f / x;
  float p = 0.39894228f + t*(0.01328592f + t*(0.00225319f + t*(-0.00157565f +
            t*(0.00916281f + t*(-0.02057706f + t*(0.02635537f +
            t*(-0.01647633f + t*0.00392377f)))))));
  return __expf(x) * p / sqrtf(x);
}

// L[l][a] = exp(-2*pi*i * l*(a-128)/512) / phi_hat(a-128), l in [0,512), a in [0,256)
__global__ void build_L(float* __restrict__ Lr, float* __restrict__ Li) {
  int l = blockIdx.x;    // 0..511
  int a = threadIdx.x;   // 0..255
  int k = a - 128;
  const float bwin = 4.712388980384690f;            // pi*(2 - 256/512) = 1.5*pi
  float wk = 3.14159265358979f * (float)k / 256.0f; // 2*pi*k/512
  float arg = bwin * bwin - wk * wk;                // > 0 for all k
  float z = 4.0f * sqrtf(arg);                      // M * sqrt(arg)
  float ph = bessel_i0_large(z) * (1.0f / 512.0f);  // phi_hat_1d
  float inv = 1.0f / ph;
  // exact integer phase reduction: angle = -2*pi*((l*k) mod 512)/512
  int mm = (l * k) % 512; if (mm < 0) mm += 512;
  float th = -3.14159265358979f * (float)mm / 256.0f;
  float s, c;
  __sincosf(th, &s, &c);
  Lr[l * 256 + a] = c * inv;
  Li[l * 256 + a] = s * inv;
}

// M1 = L (512x256 cplx) * F (256x256 cplx), per batch.  1 wave per 16x16 tile.
__global__ void gemm_LF(const float* __restrict__ Lr, const float* __restrict__ Li,
                        const float* __restrict__ Fr, const float* __restrict__ Fi,
                        float* __restrict__ M1r, float* __restrict__ M1i) {
  int wave = threadIdx.x >> 5;
  int lane = threadIdx.x & 31;
  int g = blockIdx.x * 8 + wave;       // 1024 tiles total
  int batch = g >> 9;
  int t = g & 511;                     // 32 x 16 tiles
  int m0 = (t >> 4) << 4;              // l1 base
  int n0 = (t & 15) << 4;              // a2 base
  const float* fr = Fr + batch * 65536;
  const float* fi = Fi + batch * 65536;
  int mA = lane & 15, kA = (lane >> 4) << 1;   // A frag: row, K-pair base
  int nB = lane & 15, rB = (lane >> 4) << 1;   // B frag: col, row-pair base
  v8f pr = {}, qr = {}, pi = {}, qi = {};
  for (int kk = 0; kk < 256; kk += 4) {
    int aoff = (m0 + mA) * 256 + kk + kA;       // 8B aligned (kA even)
    v2f ar = *(const v2f*)(Lr + aoff);
    v2f ai = *(const v2f*)(Li + aoff);
    int b0 = (kk + rB) * 256 + n0 + nB;
    v2f br, bi;
    br.x = fr[b0];        br.y = fr[b0 + 256];
    bi.x = fi[b0];        bi.y = fi[b0 + 256];
    pr = wmma4(ar, br, pr);   // Re*Re
    qr = wmma4(ai, bi, qr);   // Im*Im
    pi = wmma4(ar, bi, pi);   // Re*Im
    qi = wmma4(ai, br, qi);   // Im*Re
  }
  int mo = (lane >> 4) << 3;
  float* outr = M1r + batch * 131072;
  float* outi = M1i + batch * 131072;
#pragma unroll
  for (int v = 0; v < 8; v++) {
    int row = m0 + v + mo, col = n0 + nB;
    outr[row * 256 + col] = pr[v] - qr[v];
    outi[row * 256 + col] = pi[v] + qi[v];
  }
}

// Y = M1 (512x256 cplx) * L^T (256x512 cplx), per batch. 1 wave per 16x16 tile.
// Output stored INTERLEAVED (re,im) so the gather stage does one b64 per tap.
__global__ void gemm_ML(const float* __restrict__ M1r, const float* __restrict__ M1i,
                        const float* __restrict__ Lr, const float* __restrict__ Li,
                        v2f* __restrict__ Yri) {
  int wave = threadIdx.x >> 5;
  int lane = threadIdx.x & 31;
  int g = blockIdx.x * 8 + wave;       // 2048 tiles total
  int batch = g >> 10;
  int t = g & 1023;                    // 32 x 32 tiles
  int m0 = (t >> 5) << 4;              // l1 base
  int n0 = (t & 31) << 4;              // l2 base
  const float* ar_base = M1r + batch * 131072;
  const float* ai_base = M1i + batch * 131072;
  int mA = lane & 15, kA = (lane >> 4) << 1;
  int nB = lane & 15, rB = (lane >> 4) << 1;
  v8f pr = {}, qr = {}, pi = {}, qi = {};
  for (int kk = 0; kk < 256; kk += 4) {
    int aoff = (m0 + mA) * 256 + kk + kA;
    v2f ar = *(const v2f*)(ar_base + aoff);
    v2f ai = *(const v2f*)(ai_base + aoff);
    // B[r][n] = L[n0+n][kk+r]: consecutive r -> consecutive in L row, 8B aligned
    int boff = (n0 + nB) * 256 + kk + rB;
    v2f br = *(const v2f*)(Lr + boff);
    v2f bi = *(const v2f*)(Li + boff);
    pr = wmma4(ar, br, pr);
    qr = wmma4(ai, bi, qr);
    pi = wmma4(ar, bi, pi);
    qi = wmma4(ai, br, qi);
  }
  int mo = (lane >> 4) << 3;
  v2f* out = Yri + batch * 262144;
#pragma unroll
  for (int v = 0; v < 8; v++) {
    int row = m0 + v + mo, col = n0 + nB;
    v2f o; o.x = pr[v] - qr[v]; o.y = pi[v] + qi[v];
    out[row * 512 + col] = o;           // global_store_b64
  }
}

// Per-point 8x8 windowed gather: out[j] = sum gv * w1(x)w2(x)
// Y grid is 4 MB interleaved complex -> fully L2-resident; one b64 per tap.
__global__ void gather_k(const float* __restrict__ X,
                         const v2f* __restrict__ Yri,
                         v2f* __restrict__ out) {
  int j = blockIdx.x * blockDim.x + threadIdx.x;   // 0 .. B*J-1
  int b = j >> 18;                                 // J = 2^18
  float x0 = X[2 * j], x1 = X[2 * j + 1];
  float u0 = x0 * 512.0f, u1 = x1 * 512.0f;
  int l00 = (int)floorf(u0) - 3;
  int l10 = (int)floorf(u1) - 3;
  const float bwin = 4.712388980384690f;   // 1.5*pi
  float w0[8], w1[8];
  int i0a[8], i1a[8];
#pragma unroll
  for (int t = 0; t < 8; t++) {
    float s0 = u0 - (float)(l00 + t);           // n*(x - l/n)
    float a0 = 16.0f - s0 * s0;
    if (a0 > 0.0f) {
      float sq = sqrtf(a0);
      float e = __expf(bwin * sq);
      w0[t] = (e - 1.0f / e) * 0.159154943091895f / sq;  // sinh(b*sq)/(pi*sq)
    } else w0[t] = 0.0f;
    float s1 = u1 - (float)(l10 + t);
    float a1 = 16.0f - s1 * s1;
    if (a1 > 0.0f) {
      float sq = sqrtf(a1);
      float e = __expf(bwin * sq);
      w1[t] = (e - 1.0f / e) * 0.159154943091895f / sq;
    } else w1[t] = 0.0f;
    i0a[t] = (l00 + t + 512) & 511;   // gather index into standard-order DFT
    i1a[t] = (l10 + t + 512) & 511;
  }
  const v2f* y = Yri + b * 262144;
  float accr = 0.0f, acci = 0.0f;
#pragma unroll
  for (int d1 = 0; d1 < 8; d1++) {
    int base = i0a[d1] << 9;
    float wa = w0[d1];
#pragma unroll
    for (int d2 = 0; d2 < 8; d2++) {
      float w = wa * w1[d2];
      v2f g = y[base + i1a[d2]];      // global_load_b64 (L2 hit)
      accr += w * g.x;
      acci += w * g.y;
    }
  }
  v2f o; o.x = accr; o.y = acci;
  out[j] = o;                          // complex64 interleaved (re, im)
}

extern "C" void kernel_launch(void* const* d_in, const int* in_sizes, int n_in,
                              void* d_out, int out_size, void* d_ws, size_t ws_size,
                              hipStream_t stream) {
  (void)in_sizes; (void)n_in; (void)out_size; (void)ws_size;
  const float* x  = (const float*)d_in[0];   // (B, J, 2)
  const float* fr = (const float*)d_in[1];   // (B, 256, 256)
  const float* fi = (const float*)d_in[2];   // (B, 256, 256)
  float* ws = (float*)d_ws;
  float* Lr  = ws;                 // 512*256            = 131072
  float* Li  = ws + 131072;        //                     131072
  float* M1r = ws + 262144;        // B*512*256          = 262144
  float* M1i = ws + 524288;        //                     262144
  v2f*   Yri = (v2f*)(ws + 786432); // B*512*512 complex = 1048576 floats (7 MB total)

  build_L<<<512, 256, 0, stream>>>(Lr, Li);
  gemm_LF<<<128, 256, 0, stream>>>(Lr, Li, fr, fi, M1r, M1i);
  gemm_ML<<<256, 256, 0, stream>>>(M1r, M1i, Lr, Li, Yri);
  gather_k<<<2048, 256, 0, stream>>>(x, Yri, (v2f*)d_out);
}